// CurveNet_61692910239914
// MI455X (gfx1250) — compile-verified
//
#include <hip/hip_runtime.h>
#include <hip/hip_bf16.h>

// CurveNet forward for MI455X (gfx1250, wave32).
// GEMMs run on V_WMMA_F32_16X16X4_F32 (fp32-exact, matrix pipe); workload is
// HBM-bound (~300MB intermediates @ 23.3 TB/s >> ~20 GFLOP of math), so fp32
// WMMA is the right precision choice -- no accuracy tradeoff, no benefit from f16.

typedef __attribute__((ext_vector_type(2))) float v2f;
typedef __attribute__((ext_vector_type(8))) float v8f;

#define TPB 256

// ---------------------------------------------------------------------------
// WMMA GEMM:  D[r,c] = act( (sum_k A[r,k]*W[k,c]) * g[c] + b[c] )
// A: (M,Kd) row-major, W: (Kd,Nc) row-major. One wave32 per 16x16 output tile.
// ISA layouts (CDNA5 ISA):
//   A 16x4 f32 : lane L holds A[M=L%16][K = 2*(L/16)+j] in vgpr j (j=0,1)
//   B  4x16 f32: lane L holds W[K = 2*(L/16)+j][N=L%16]  in vgpr j
//   C/D 16x16  : vgpr r = C[M = r + 8*(L/16)][N = L%16]
// Main k-loop is branch-free (all guards hoisted); EXEC stays all-ones as the
// ISA requires for WMMA. Prefetch is speculative -- safe past the end.
// ---------------------------------------------------------------------------
__global__ void gemm_affine_wmma(const float* __restrict__ A,
                                 const float* __restrict__ W,
                                 const float* __restrict__ g,
                                 const float* __restrict__ b,
                                 float* __restrict__ D,
                                 int M, int Kd, int Nc, int doRelu)
{
    const int lane = threadIdx.x & 31;
    const int wave = threadIdx.x >> 5;
    const int tile = blockIdx.x * (blockDim.x >> 5) + wave;
    const int tilesN = Nc >> 4;
    const int tM = (tile / tilesN) << 4;
    const int tN = (tile % tilesN) << 4;
    if (tM >= M) return;                 // wave-uniform: EXEC stays all-ones

    const int half = lane >> 4;          // 0: lanes 0-15, 1: lanes 16-31
    const int l16  = lane & 15;
    const int rowA = tM + l16;
    const int colB = tN + l16;

    const float* __restrict__ Arow = A + (size_t)rowA * Kd + (half << 1);
    const float* __restrict__ Wcol = W + (size_t)(half << 1) * Nc + colB;
    const size_t w4 = (size_t)4 * Nc;    // W advance per k-step of 4

    v8f acc = (v8f){0.f, 0.f, 0.f, 0.f, 0.f, 0.f, 0.f, 0.f};

    const int kMain = Kd & ~3;
    int k0 = 0;

    // -------- unrolled-by-2 branch-free main loop (8 loads + 2 WMMA) --------
    for (; k0 + 8 <= kMain; k0 += 8) {
        v2f a0, b0, a1, b1;
        a0.x = Arow[0];  a0.y = Arow[1];
        b0.x = Wcol[0];  b0.y = Wcol[Nc];
        a1.x = Arow[4];  a1.y = Arow[5];
        b1.x = Wcol[w4]; b1.y = Wcol[w4 + Nc];
        __builtin_prefetch(Arow + 8, 0, 1);          // global_prefetch_b8
        __builtin_prefetch(Wcol + 2 * w4, 0, 1);     // speculative: OOB-safe
        acc = __builtin_amdgcn_wmma_f32_16x16x4_f32(false, a0, false, b0,
                                                    (short)0, acc, false, false);
        acc = __builtin_amdgcn_wmma_f32_16x16x4_f32(false, a1, false, b1,
                                                    (short)0, acc, false, false);
        Arow += 8;
        Wcol += 2 * w4;
    }
    // -------- remaining single 4-step (still branch-free) --------
    for (; k0 < kMain; k0 += 4) {
        v2f a0, b0;
        a0.x = Arow[0]; a0.y = Arow[1];
        b0.x = Wcol[0]; b0.y = Wcol[Nc];
        acc = __builtin_amdgcn_wmma_f32_16x16x4_f32(false, a0, false, b0,
                                                    (short)0, acc, false, false);
        Arow += 4;
        Wcol += w4;
    }
    // -------- tail for Kd % 4 != 0 (executed at most once; e.g. Kd = 9) -----
    if (k0 < Kd) {
        const int ka = k0 + (half << 1);
        v2f a0, b0;
        a0.x = (ka     < Kd) ? Arow[0]  : 0.f;
        a0.y = (ka + 1 < Kd) ? Arow[1]  : 0.f;
        b0.x = (ka     < Kd) ? Wcol[0]  : 0.f;
        b0.y = (ka + 1 < Kd) ? Wcol[Nc] : 0.f;
        acc = __builtin_amdgcn_wmma_f32_16x16x4_f32(false, a0, false, b0,
                                                    (short)0, acc, false, false);
    }

    const float gc = g[colB];
    const float bc = b[colB];
#pragma unroll
    for (int r = 0; r < 8; ++r) {
        const int row = tM + (half << 3) + r;
        float v = acc[r] * gc + bc;
        if (doRelu) v = fmaxf(v, 0.f);
        if (row < M) D[(size_t)row * Nc + colB] = v;
    }
}

// ---------------------------------------------------------------------------
// (B,3,N) -> (B,N,3)
// ---------------------------------------------------------------------------
__global__ void transpose_xyz_kernel(const float* __restrict__ xyz,
                                     float* __restrict__ pts, int B, int N)
{
    int e = blockIdx.x * blockDim.x + threadIdx.x;
    int total = B * N * 3;
    if (e >= total) return;
    int j = e % 3; int r = e / 3;
    int i = r % N; int b = r / N;
    pts[e] = xyz[((size_t)b * 3 + j) * N + i];
}

// ---------------------------------------------------------------------------
// Farthest point sampling; one block per batch, LDS distance array + argmax.
// Replicates the scan semantics: out[t] = incoming "far" (out[0] = 0).
// ---------------------------------------------------------------------------
__global__ void fps_kernel(const float* __restrict__ xyz,  // (B,n,3)
                           int* __restrict__ out,          // (B,npoint)
                           int n, int npoint)
{
    extern __shared__ float smem[];
    float* dist = smem;                       // n
    float* rv   = smem + n;                   // blockDim
    int*   ri   = (int*)(rv + blockDim.x);    // blockDim

    const int b = blockIdx.x;
    const float* P = xyz + (size_t)b * n * 3;
    int* O = out + (size_t)b * npoint;

    for (int i = threadIdx.x; i < n; i += blockDim.x) dist[i] = 1e10f;
    __syncthreads();

    int far = 0;
    for (int t = 0; t < npoint; ++t) {
        if (threadIdx.x == 0) O[t] = far;
        const float sx = P[far * 3], sy = P[far * 3 + 1], sz = P[far * 3 + 2];
        float bestv = -1.f; int besti = 0;
        for (int i = threadIdx.x; i < n; i += blockDim.x) {
            float dx = P[i * 3] - sx, dy = P[i * 3 + 1] - sy, dz = P[i * 3 + 2] - sz;
            float d  = fminf(dist[i], dx * dx + dy * dy + dz * dz);
            dist[i] = d;
            if (d > bestv) { bestv = d; besti = i; }
        }
        rv[threadIdx.x] = bestv; ri[threadIdx.x] = besti;
        __syncthreads();
        for (int s = blockDim.x >> 1; s > 0; s >>= 1) {
            if (threadIdx.x < s) {
                float ov = rv[threadIdx.x + s]; int oi = ri[threadIdx.x + s];
                if (ov > rv[threadIdx.x] ||
                    (ov == rv[threadIdx.x] && oi < ri[threadIdx.x])) {
                    rv[threadIdx.x] = ov; ri[threadIdx.x] = oi;
                }
            }
            __syncthreads();
        }
        far = ri[0];
        __syncthreads();
    }
}

// ---------------------------------------------------------------------------
// KNN: one thread per query, register insertion-sort top-KSEL (ascending d).
// ---------------------------------------------------------------------------
template <int KSEL>
__global__ void knn_kernel(const float* __restrict__ q,   // (B,nq,3)
                           const float* __restrict__ ref, // (B,nr,3)
                           int* __restrict__ outIdx,      // (B,nq,KSEL)
                           int nq, int nr)
{
    const int b = blockIdx.y;
    const int i = blockIdx.x * blockDim.x + threadIdx.x;
    if (i >= nq) return;
    const float* Q = q + ((size_t)b * nq + i) * 3;
    const float qx = Q[0], qy = Q[1], qz = Q[2];
    float bd[KSEL]; int bi[KSEL];
#pragma unroll
    for (int t = 0; t < KSEL; ++t) { bd[t] = 3.4028235e38f; bi[t] = 0; }
    const float* R = ref + (size_t)b * nr * 3;
    for (int j = 0; j < nr; ++j) {
        const float dx = qx - R[j * 3], dy = qy - R[j * 3 + 1], dz = qz - R[j * 3 + 2];
        const float d = dx * dx + dy * dy + dz * dz;
        if (d < bd[KSEL - 1]) {
            int t = KSEL - 1;
            while (t > 0 && bd[t - 1] > d) { bd[t] = bd[t - 1]; bi[t] = bi[t - 1]; --t; }
            bd[t] = d; bi[t] = j;
        }
    }
    int* O = outIdx + ((size_t)b * nq + i) * KSEL;
#pragma unroll
    for (int t = 0; t < KSEL; ++t) O[t] = bi[t];
}

// ---------------------------------------------------------------------------
// pos_encoding: feat[b,i,k,:] = [ctr(3), nbr(3), nbr-ctr(3)]
// ---------------------------------------------------------------------------
__global__ void pos_encoding_kernel(const float* __restrict__ xyz,  // (B,n,3)
                                    const int* __restrict__ idx,    // (B,n,K)
                                    float* __restrict__ feat,       // (B,n,K,9)
                                    int B, int n, int Kn)
{
    int e = blockIdx.x * blockDim.x + threadIdx.x;
    int total = B * n * Kn;
    if (e >= total) return;
    int r = e / Kn;
    int i = r % n; int b = r / n;
    const float* C = xyz + ((size_t)b * n + i) * 3;
    const int j = idx[e];
    const float* Nb = xyz + ((size_t)b * n + j) * 3;
    float* f = feat + (size_t)e * 9;
    f[0] = C[0];  f[1] = C[1];  f[2] = C[2];
    f[3] = Nb[0]; f[4] = Nb[1]; f[5] = Nb[2];
    f[6] = Nb[0] - C[0]; f[7] = Nb[1] - C[1]; f[8] = Nb[2] - C[2];
}

// ---------------------------------------------------------------------------
// gather rows:  dst[b,r,c] = src[b, idx[b,r], c]
// ---------------------------------------------------------------------------
__global__ void gather_rows_kernel(const float* __restrict__ src,  // (B,n,C)
                                   const int* __restrict__ idx,    // (B,m)
                                   float* __restrict__ dst,        // (B,m,C)
                                   int B, int n, int m, int C)
{
    int e = blockIdx.x * blockDim.x + threadIdx.x;
    int total = B * m * C;
    if (e >= total) return;
    int c = e % C; int r = e / C;
    int i = r % m; int b = r / m;
    dst[e] = src[((size_t)b * n + idx[(size_t)b * m + i]) * C + c];
}

// ---------------------------------------------------------------------------
// tmp[b,i,k,c] = h[b, idx[b,i,k], c] + pe[b,i,k,c]
// ---------------------------------------------------------------------------
__global__ void gather_add_kernel(const float* __restrict__ h,   // (B,n,C)
                                  const int* __restrict__ idx,   // (B,n,K)
                                  const float* __restrict__ pe,  // (B,n,K,C)
                                  float* __restrict__ tmp,       // (B,n,K,C)
                                  int B, int n, int Kn, int C)
{
    int e = blockIdx.x * blockDim.x + threadIdx.x;
    int total = B * n * Kn * C;
    if (e >= total) return;
    int c = e % C; int r = e / C;
    int b = r / (n * Kn);
    int j = idx[r];
    tmp[e] = h[((size_t)b * n + j) * C + c] + pe[e];
}

// ---------------------------------------------------------------------------
// out[b,i,c] = max_k e[b,i,k,c]
// ---------------------------------------------------------------------------
__global__ void max_over_k_kernel(const float* __restrict__ ein,  // (B,n,K,C)
                                  float* __restrict__ out,        // (B,n,C)
                                  int B, int n, int Kn, int C)
{
    int e = blockIdx.x * blockDim.x + threadIdx.x;
    int total = B * n * C;
    if (e >= total) return;
    int c = e % C; int r = e / C;
    const float* base = ein + ((size_t)r * Kn) * C + c;
    float m = base[0];
    for (int k = 1; k < Kn; ++k) m = fmaxf(m, base[(size_t)k * C]);
    out[e] = m;
}

// ---------------------------------------------------------------------------
// x = relu(a + b)
// ---------------------------------------------------------------------------
__global__ void add_relu_kernel(const float* __restrict__ a,
                                const float* __restrict__ b,
                                float* __restrict__ x, int total)
{
    int e = blockIdx.x * blockDim.x + threadIdx.x;
    if (e >= total) return;
    x[e] = fmaxf(a[e] + b[e], 0.f);
}

// ---------------------------------------------------------------------------
// head: out[b, c] = max_j h4[b,j,c]; out[b, 1024+c] = mean_j h4[b,j,c]
// h4: (B, 64, 1024)
// ---------------------------------------------------------------------------
__global__ void head_reduce_kernel(const float* __restrict__ h4,
                                   float* __restrict__ out, int B)
{
    int e = blockIdx.x * blockDim.x + threadIdx.x;
    if (e >= B * 1024) return;
    int c = e % 1024, b = e / 1024;
    float mx = -3.4028235e38f, sm = 0.f;
    for (int j = 0; j < 64; ++j) {
        float v = h4[((size_t)b * 64 + j) * 1024 + c];
        mx = fmaxf(mx, v); sm += v;
    }
    out[(size_t)b * 2048 + c] = mx;
    out[(size_t)b * 2048 + 1024 + c] = sm * (1.f / 64.f);
}

// ---------------------------------------------------------------------------
// sample_and_group output: pcl (B,128,32,67) = [g_xyz(3) | g_pts(64)]
// pc: (B,1024,3) = raw xyz buffer reinterpreted (reference reshapes, no transpose)
// feats[b,r,j] = h4[b,j,r]  (x_t = swapaxes(h,1,2))
// ---------------------------------------------------------------------------
__global__ void sg_kernel(const float* __restrict__ pc,      // (B,1024,3)
                          const float* __restrict__ newxyz,  // (B,128,3)
                          const int* __restrict__ idx,       // (B,128,32)
                          const float* __restrict__ h4,      // (B,64,1024)
                          float* __restrict__ pcl, int B)
{
    int e = blockIdx.x * blockDim.x + threadIdx.x;
    int total = B * 128 * 32 * 67;
    if (e >= total) return;
    int c = e % 67; int r = e / 67;
    int k = r % 32; r /= 32;
    int i = r % 128; int b = r / 128;
    const int jv = idx[((size_t)b * 128 + i) * 32 + k];
    float v;
    if (c < 3) {
        v = pc[((size_t)b * 1024 + jv) * 3 + c] -
            newxyz[((size_t)b * 128 + i) * 3 + c];
    } else {
        v = h4[((size_t)b * 64 + (c - 3)) * 1024 + jv];
    }
    pcl[e] = v;
}

// ===========================================================================
// Host orchestration
// ===========================================================================
extern "C" void kernel_launch(void* const* d_in, const int* in_sizes, int n_in,
                              void* d_out, int out_size, void* d_ws, size_t ws_size,
                              hipStream_t stream)
{
    (void)in_sizes; (void)n_in; (void)out_size; (void)ws_size;

    constexpr int B = 16, N0 = 1024, KNN = 20;
    const int npoints[8] = {1024, 1024, 1024, 1024, 256, 256, 64, 64};
    const int cfg[8][3]  = {{32,64,32},{64,64,16},{64,128,64},{128,128,32},
                            {128,256,128},{256,256,64},{256,512,256},{512,512,128}};

    // --- input pointers (jax tree_flatten of params dict: keys sorted) ---
    // 0: xyz; 1: W0; 2: b0; 3+15s+j: cics[s] keys
    // [W1,W2,W3,Wp,Ws,b1,b2,b3,bp,bs,g1,g2,g3,gp,gs]; 123: g0; 124-126: lpfa W,b,g
    auto F = [&](int i) { return (const float*)d_in[i]; };
    const float* xyz_in = F(0);
    const float* W0p = F(1);  const float* b0p = F(2);  const float* g0p = F(123);
    const float* lpW = F(124); const float* lpB = F(125); const float* lpG = F(126);
    auto CW1 = [&](int s){ return F(3 + s*15 + 0);  };
    auto CW2 = [&](int s){ return F(3 + s*15 + 1);  };
    auto CW3 = [&](int s){ return F(3 + s*15 + 2);  };
    auto CWp = [&](int s){ return F(3 + s*15 + 3);  };
    auto CWs = [&](int s){ return F(3 + s*15 + 4);  };
    auto Cb1 = [&](int s){ return F(3 + s*15 + 5);  };
    auto Cb2 = [&](int s){ return F(3 + s*15 + 6);  };
    auto Cb3 = [&](int s){ return F(3 + s*15 + 7);  };
    auto Cbp = [&](int s){ return F(3 + s*15 + 8);  };
    auto Cbs = [&](int s){ return F(3 + s*15 + 9);  };
    auto Cg1 = [&](int s){ return F(3 + s*15 + 10); };
    auto Cg2 = [&](int s){ return F(3 + s*15 + 11); };
    auto Cg3 = [&](int s){ return F(3 + s*15 + 12); };
    auto Cgp = [&](int s){ return F(3 + s*15 + 13); };
    auto Cgs = [&](int s){ return F(3 + s*15 + 14); };

    // --- workspace bump allocator ---
    float* ws = (float*)d_ws;
    size_t off = 0;
    auto alloc = [&](size_t nf) { float* p = ws + off; off += nf; return p; };

    float* pts    = alloc((size_t)B * N0 * 3);
    float* xyzA   = alloc((size_t)B * N0 * 3);
    float* xyzB   = alloc((size_t)B * N0 * 3);
    float* xA     = alloc((size_t)B * 1024 * 128);
    float* xB     = alloc((size_t)B * 1024 * 128);
    int*   idxBuf = (int*)alloc((size_t)B * 1024 * 32);
    int*   fidx   = (int*)alloc((size_t)B * 1024);
    float* feat9  = alloc((size_t)B * 1024 * KNN * 9);
    float* hbuf   = alloc((size_t)B * 65536);          // max n*cmid
    float* pebuf  = alloc((size_t)B * 1024 * KNN * 64);// max n*K*cmid
    float* tmpbuf = alloc((size_t)B * 1024 * KNN * 64);
    float* ebuf   = alloc((size_t)B * 1024 * KNN * 64);
    float* aggbuf = alloc((size_t)B * 65536);
    float* out3   = alloc((size_t)B * 131072);         // max n*cout
    float* scbuf  = alloc((size_t)B * 131072);
    float* h4     = alloc((size_t)B * 64 * 1024);
    float* newxyz = alloc((size_t)B * 128 * 3);

    auto nblk = [](long long total) { return (int)((total + TPB - 1) / TPB); };

    auto gemm = [&](const float* A, const float* W, const float* g,
                    const float* b, float* D, int M, int Kd, int Nc, int relu) {
        long long tiles = (long long)((M + 15) / 16) * (Nc / 16);
        int waves = TPB / 32;
        gemm_affine_wmma<<<(int)((tiles + waves - 1) / waves), TPB, 0, stream>>>(
            A, W, g, b, D, M, Kd, Nc, relu);
    };

    // ---- pts = transpose(xyz) ----
    transpose_xyz_kernel<<<nblk((long long)B * N0 * 3), TPB, 0, stream>>>(
        xyz_in, pts, B, N0);

    // ---- lpfa_initial ----
    {
        dim3 g(nblk(N0), B);
        knn_kernel<KNN><<<g, TPB, 0, stream>>>(pts, pts, idxBuf, N0, N0);
        pos_encoding_kernel<<<nblk((long long)B * N0 * KNN), TPB, 0, stream>>>(
            pts, idxBuf, feat9, B, N0, KNN);
        gemm(feat9, lpW, lpG, lpB, tmpbuf, B * N0 * KNN, 9, 32, 1);
        max_over_k_kernel<<<nblk((long long)B * N0 * 32), TPB, 0, stream>>>(
            tmpbuf, xA, B, N0, KNN, 32);
    }

    // ---- CIC stages ----
    float* xbufs[2]   = {xA, xB};
    float* xyzbufs[2] = {xyzA, xyzB};
    float* curX   = xA;
    float* curXyz = pts;
    int n = N0;

    for (int s = 0; s < 8; ++s) {
        const int cin = cfg[s][0], cout = cfg[s][1], cmid = cfg[s][2];
        const int npt = npoints[s];

        if (npt < n) {
            size_t shmem = (size_t)(n + 2 * TPB) * sizeof(float);
            fps_kernel<<<B, TPB, shmem, stream>>>(curXyz, fidx, n, npt);
            float* dstXyz = (curXyz == xyzbufs[0]) ? xyzbufs[1] : xyzbufs[0];
            float* dstX   = (curX   == xbufs[0])   ? xbufs[1]   : xbufs[0];
            gather_rows_kernel<<<nblk((long long)B * npt * 3), TPB, 0, stream>>>(
                curXyz, fidx, dstXyz, B, n, npt, 3);
            gather_rows_kernel<<<nblk((long long)B * npt * cin), TPB, 0, stream>>>(
                curX, fidx, dstX, B, n, npt, cin);
            curXyz = dstXyz; curX = dstX; n = npt;
        }

        // h = relu(x @ W1 * g1 + b1)
        gemm(curX, CW1(s), Cg1(s), Cb1(s), hbuf, B * n, cin, cmid, 1);
        // knn on current xyz
        dim3 g(nblk(n), B);
        knn_kernel<KNN><<<g, TPB, 0, stream>>>(curXyz, curXyz, idxBuf, n, n);
        // pe = pos_encoding @ Wp * gp + bp
        pos_encoding_kernel<<<nblk((long long)B * n * KNN), TPB, 0, stream>>>(
            curXyz, idxBuf, feat9, B, n, KNN);
        gemm(feat9, CWp(s), Cgp(s), Cbp(s), pebuf, B * n * KNN, 9, cmid, 0);
        // tmp = gather(h, idx) + pe
        gather_add_kernel<<<nblk((long long)B * n * KNN * cmid), TPB, 0, stream>>>(
            hbuf, idxBuf, pebuf, tmpbuf, B, n, KNN, cmid);
        // e = relu(tmp @ W2 * g2 + b2)
        gemm(tmpbuf, CW2(s), Cg2(s), Cb2(s), ebuf, B * n * KNN, cmid, cmid, 1);
        // agg = max_k e
        max_over_k_kernel<<<nblk((long long)B * n * cmid), TPB, 0, stream>>>(
            ebuf, aggbuf, B, n, KNN, cmid);
        // out = agg @ W3 * g3 + b3 ; sc = x @ Ws * gs + bs
        gemm(aggbuf, CW3(s), Cg3(s), Cb3(s), out3, B * n, cmid, cout, 0);
        gemm(curX, CWs(s), Cgs(s), Cbs(s), scbuf, B * n, cin, cout, 0);
        // x = relu(out + sc)
        float* xNext = (curX == xbufs[0]) ? xbufs[1] : xbufs[0];
        add_relu_kernel<<<nblk((long long)B * n * cout), TPB, 0, stream>>>(
            out3, scbuf, xNext, B * n * cout);
        curX = xNext;
    }

    // ---- head: h4 = relu(x @ W0 * g0 + b0), (B,64,1024) ----
    gemm(curX, W0p, g0p, b0p, h4, B * 64, 512, 1024, 1);
    head_reduce_kernel<<<nblk((long long)B * 1024), TPB, 0, stream>>>(
        h4, (float*)d_out, B);

    // ---- sample_and_group on pc = raw xyz buffer viewed (B,1024,3) ----
    {
        const float* pc = xyz_in;  // reference does reshape, not transpose
        size_t shmem = (size_t)(N0 + 2 * TPB) * sizeof(float);
        fps_kernel<<<B, TPB, shmem, stream>>>(pc, fidx, N0, 128);
        gather_rows_kernel<<<nblk((long long)B * 128 * 3), TPB, 0, stream>>>(
            pc, fidx, newxyz, B, N0, 128, 3);
        dim3 g(nblk(128), B);
        knn_kernel<32><<<g, TPB, 0, stream>>>(newxyz, pc, idxBuf, 128, N0);
        float* pcl = (float*)d_out + (size_t)B * 2048;
        sg_kernel<<<nblk((long long)B * 128 * 32 * 67), TPB, 0, stream>>>(
            pc, newxyz, idxBuf, h4, pcl, B);
    }
}